// NodeFeatures_48473000903548
// MI455X (gfx1250) — compile-verified
//
#include <hip/hip_runtime.h>
#include <hip/hip_bf16.h>

// ---------------------------------------------------------------------------
// Problem: B=8, N=256, H=128
//   Ux = x @ U_w^T + U_b            (2048x128x128 f32 GEMM)  -> WMMA f32 16x16x4
//   Vx = x @ V_w^T + V_b            (2048x128x128 f32 GEMM)  -> WMMA f32 16x16x4
//   out[b,i,h] = Ux[b,i,h] + (sum_j eg[b,i,j,h]*Vx[b,j,h]) / (sum_j eg[b,i,j,h] + 1e-20)
// edge_gate is 256MB read-once -> bandwidth bound (~11us @ 23.3 TB/s); stream NT.
// ---------------------------------------------------------------------------

typedef float v2f __attribute__((ext_vector_type(2)));
typedef float v4f __attribute__((ext_vector_type(4)));
typedef float v8f __attribute__((ext_vector_type(8)));

#define H 128
#define NN 256
#define BB 8
#define M_TOTAL (BB * NN)   // 2048 rows

// ---------------------------------------------------------------------------
// Kernel 1: both GEMMs (U and V) via V_WMMA_F32_16X16X4_F32.
// Wave-per-tile: 2048 tiles total = 2 matrices * 128 m-tiles * 8 n-tiles.
// A (16x4, MxK): lane 0-15 -> M=lane, VGPR v -> K = v + 2*(lane>>4)
// B (4x16, KxN): mirrored; B[k][n] = W[n][k], so lane indexes W row n.
// C/D (16x16):   VGPR r -> M = r + 8*(lane>>4), N = lane&15
// ---------------------------------------------------------------------------
__global__ __launch_bounds__(256)
void nf_gemm_uv(const float* __restrict__ x,
                const float* __restrict__ U_w, const float* __restrict__ U_b,
                const float* __restrict__ V_w, const float* __restrict__ V_b,
                float* __restrict__ Ux_out, float* __restrict__ Vx_out)
{
    const int lane = threadIdx.x & 31;
    const int wave = threadIdx.x >> 5;
    const int gt   = blockIdx.x * 8 + wave;   // 0..2047 (uniform per wave)

    const int which  = gt >> 10;              // 0 = U, 1 = V
    const int rem    = gt & 1023;
    const int m_base = (rem >> 3) << 4;       // 0..2032 step 16
    const int n_base = (rem & 7) << 4;        // 0..112  step 16

    const float* __restrict__ W    = which ? V_w : U_w;
    const float* __restrict__ bias = which ? V_b : U_b;
    float* __restrict__ dst        = which ? Vx_out : Ux_out;

    const int row_a = m_base + (lane & 15);   // x row (M)
    const int row_w = n_base + (lane & 15);   // weight row = output feature (N)
    const int koff  = (lane >> 4) << 1;       // 0 or 2

    const float* pa = x + row_a * H + koff;
    const float* pw = W + row_w * H + koff;

    v8f c = {0.f, 0.f, 0.f, 0.f, 0.f, 0.f, 0.f, 0.f};

#if __has_builtin(__builtin_amdgcn_wmma_f32_16x16x4_f32)
    #pragma unroll
    for (int k0 = 0; k0 < H; k0 += 4) {
        v2f a = *(const v2f*)(pa + k0);
        v2f b = *(const v2f*)(pw + k0);
        c = __builtin_amdgcn_wmma_f32_16x16x4_f32(
                /*neg_a=*/false, a, /*neg_b=*/false, b,
                /*c_mod=*/(short)0, c, /*reuse_a=*/false, /*reuse_b=*/false);
    }
    const int n  = n_base + (lane & 15);
    const int m0 = m_base + ((lane >> 4) << 3);
    const float bn = bias[n];
    #pragma unroll
    for (int r = 0; r < 8; ++r)
        dst[(m0 + r) * H + n] = c[r] + bn;
#else
    // Scalar fallback (same output mapping) — only if the f32 WMMA builtin is absent.
    const int n  = n_base + (lane & 15);
    const int m0 = m_base + ((lane >> 4) << 3);
    const float bn = bias[n];
    (void)pa; (void)pw;
    #pragma unroll
    for (int r = 0; r < 8; ++r) {
        const float* xr = x + (m0 + r) * H;
        const float* wr = W + n * H;
        float acc = 0.f;
        for (int k = 0; k < H; ++k) acc += xr[k] * wr[k];
        dst[(m0 + r) * H + n] = acc + bn;
    }
#endif
}

// ---------------------------------------------------------------------------
// Kernel 2: gated mean aggregation, fused with Ux (already in d_out).
// One block per (b,i). 256 threads = 8 j-groups x 32 h-quads.
// edge_gate row streamed once via non-temporal b128 loads (no reuse, 256MB);
// Vx[b] (128KB) hits cache across the 256 i-blocks of the same batch.
// ---------------------------------------------------------------------------
__global__ __launch_bounds__(256)
void nf_aggregate(const float* __restrict__ eg,
                  const float* __restrict__ Vx,
                  float* __restrict__ out)   // holds Ux, updated in place
{
    __shared__ v4f s_num[8][32];
    __shared__ v4f s_den[8][32];

    const int bi = blockIdx.x;        // b*256 + i, 0..2047
    const int b  = bi >> 8;
    const int hv = threadIdx.x & 31;  // h-quad: h = 4*hv .. 4*hv+3
    const int g  = threadIdx.x >> 5;  // j-group 0..7

    const v4f* egp = (const v4f*)(eg + (size_t)bi * NN * H);
    const v4f* vxp = (const v4f*)(Vx + (size_t)b  * NN * H);

    v4f num = {0.f, 0.f, 0.f, 0.f};
    v4f den = {0.f, 0.f, 0.f, 0.f};

    for (int j = g; j < NN; j += 8) {
        v4f e = __builtin_nontemporal_load(egp + j * (H / 4) + hv);
        v4f v = vxp[j * (H / 4) + hv];
        num.x += e.x * v.x;  num.y += e.y * v.y;
        num.z += e.z * v.z;  num.w += e.w * v.w;
        den.x += e.x;        den.y += e.y;
        den.z += e.z;        den.w += e.w;
    }

    s_num[g][hv] = num;
    s_den[g][hv] = den;
    __syncthreads();

    if (threadIdx.x < 32) {
        v4f tn = s_num[0][hv];
        v4f td = s_den[0][hv];
        #pragma unroll
        for (int q = 1; q < 8; ++q) {
            v4f n2 = s_num[q][hv], d2 = s_den[q][hv];
            tn.x += n2.x; tn.y += n2.y; tn.z += n2.z; tn.w += n2.w;
            td.x += d2.x; td.y += d2.y; td.z += d2.z; td.w += d2.w;
        }
        v4f* op = (v4f*)(out + (size_t)bi * H) + hv;
        v4f u = *op;
        v4f r;
        r.x = u.x + tn.x / (td.x + 1e-20f);
        r.y = u.y + tn.y / (td.y + 1e-20f);
        r.z = u.z + tn.z / (td.z + 1e-20f);
        r.w = u.w + tn.w / (td.w + 1e-20f);
        *op = r;
    }
}

// ---------------------------------------------------------------------------
extern "C" void kernel_launch(void* const* d_in, const int* in_sizes, int n_in,
                              void* d_out, int out_size, void* d_ws, size_t ws_size,
                              hipStream_t stream)
{
    const float* x   = (const float*)d_in[0];  // [8,256,128]
    const float* eg  = (const float*)d_in[1];  // [8,256,256,128]
    const float* U_w = (const float*)d_in[2];  // [128,128]
    const float* U_b = (const float*)d_in[3];  // [128]
    const float* V_w = (const float*)d_in[4];  // [128,128]
    const float* V_b = (const float*)d_in[5];  // [128]

    float* out = (float*)d_out;                // [8,256,128] -> gets Ux, then += agg
    float* Vx  = (float*)d_ws;                 // 2048*128 floats = 1 MB scratch

    // 2048 wave-tiles / 8 waves per block = 256 blocks
    nf_gemm_uv<<<256, 256, 0, stream>>>(x, U_w, U_b, V_w, V_b, out, Vx);

    // one block per (b,i)
    nf_aggregate<<<M_TOTAL, 256, 0, stream>>>(eg, Vx, out);
}